// BoxConv2d_41472204210999
// MI455X (gfx1250) — compile-verified
//
#include <hip/hip_runtime.h>
#include <cstdint>
#include <cstddef>

// Problem constants (fixed by setup_inputs: B=8, C=32, F=8, H=W=128)
#define Hd 128
#define Wd 128
#define Bn 8
#define Cn 32
#define Fn 8
#define LDSTR 132                 // LDS row stride in floats; 132 mod 64 = 4 -> conflict-friendly
#define AOFF 0                    // region A: X, then padded integral I (129 x LDSTR)
#define BOFF (129 * LDSTR)        // region B: T (row-cumsum), then E (col-interp) (129 x LDSTR)
#define LDS_FLOATS (2 * 129 * LDSTR)

typedef float v2f __attribute__((ext_vector_type(2)));
typedef float v8f __attribute__((ext_vector_type(8)));
typedef unsigned int v4u __attribute__((ext_vector_type(4)));
typedef int v8i __attribute__((ext_vector_type(8)));
typedef int v4i __attribute__((ext_vector_type(4)));

#if defined(__gfx1250__) && __has_builtin(__builtin_amdgcn_wmma_f32_16x16x4_f32)
#define HAVE_WMMA 1
#else
#define HAVE_WMMA 0
#endif

#ifndef TRY_TDM
#define TRY_TDM 1
#endif
#if TRY_TDM && defined(__gfx1250__) && __has_builtin(__builtin_amdgcn_tensor_load_to_lds) && \
    __has_builtin(__builtin_amdgcn_s_wait_tensorcnt)
#define HAVE_TDM 1
#else
#define HAVE_TDM 0
#endif

__device__ __forceinline__ float clip128(float x) { return fminf(fmaxf(x, 0.0f), 128.0f); }
__device__ __forceinline__ int imin(int a, int b) { return a < b ? a : b; }

__global__ void __launch_bounds__(256)
boxconv_fused(const float* __restrict__ x,
              const float* __restrict__ x_min, const float* __restrict__ x_max,
              const float* __restrict__ y_min, const float* __restrict__ y_max,
              float* __restrict__ out)
{
    extern __shared__ float lds[];
    const int tid = threadIdx.x;
    const int bc  = blockIdx.x;              // bc = b*Cn + c, grid = Bn*Cn = 256
    const int c   = bc & (Cn - 1);
    const float* __restrict__ xg = x + (size_t)bc * (Hd * Wd);

    // ---- Phase 0: zero border row/col of integral region; stage X into region-A interior ----
    if (tid < 129) { lds[AOFF + tid] = 0.0f; lds[AOFF + tid * LDSTR] = 0.0f; }

#if HAVE_TDM
    if ((tid >> 5) == 0) {
        // One TDM DMA: 128x128 f32 tile, global stride 128 -> LDS starting at interior
        // (byte 532 = (1*LDSTR+1)*4; dynamic LDS base is 0, kernel has no static LDS),
        // with pad_enable inserting 4 dwords after every 128 dwords -> LDS stride 132.
        uint64_t ga = (uint64_t)(uintptr_t)xg;
        v4u g0;
        g0.x = 1u;                                                 // count = 1 (valid descriptor)
        g0.y = (unsigned)((1 * LDSTR + 1) * 4);                    // lds_addr (bytes)
        g0.z = (unsigned)(ga & 0xFFFFFFFFu);                       // global_addr[31:0]
        g0.w = (unsigned)((ga >> 32) & 0x01FFFFFFu) | (2u << 30);  // global_addr[56:32] | type=2
        v8i g1;
        g1[0] = (2 << 16)        // data_size = 4 bytes
              | (1 << 20)        // pad_enable
              | (6 << 22)        // pad_interval code 6 = every 128 dwords
              | (3 << 25);       // pad_amount code 3 = 4 dwords
        g1[1] = (Wd << 16);      // tensor_dim0[15:0] = 128 (bits 79:48)
        g1[2] = (Hd << 16);      // tensor_dim0 hi = 0 | tensor_dim1[15:0] = 128
        g1[3] = (Wd << 16);      // tensor_dim1 hi = 0 | tile_dim0 = 128
        g1[4] = Hd;              // tile_dim1 = 128, tile_dim2 = 0
        g1[5] = Wd;              // tensor_dim0_stride = 128
        g1[6] = (int)(((Hd * Wd) & 0xFFFF) << 16);  // dim0_stride hi=0 | dim1_stride lo
        g1[7] = (Hd * Wd) >> 16;                    // dim1_stride hi
        v4i gz4 = {0, 0, 0, 0};
        v8i gz8 = {0, 0, 0, 0, 0, 0, 0, 0};
        __builtin_amdgcn_tensor_load_to_lds(g0, g1, gz4, gz4, gz8, 0);
        __builtin_amdgcn_s_wait_tensorcnt(0);
    }
#else
    for (int e = tid; e < Hd * Wd; e += 256) {
        const int hh = e >> 7, ww = e & 127;
        lds[AOFF + (hh + 1) * LDSTR + (ww + 1)] = xg[e];
    }
#endif
    __syncthreads();

    // ---- Phase 1+2: integral image via triangular GEMMs on WMMA f32 16x16x4 ----
#if HAVE_WMMA
    {
        const int lane = tid & 31;
        const int wv   = tid >> 5;            // 8 waves
        const int lm   = lane & 15;
        const int kh   = (lane >> 4) << 1;    // 0 or 2 : K-half select per fragment layout
        // GEMM-1: T = L * X  (row-cumsum over H). A = L analytic, B = X from LDS.
        for (int t8 = 0; t8 < 8; ++t8) {
            const int mt = t8, nt = wv;       // work ~ mt -> balanced across waves
            const int m = mt * 16 + lm;
            const int n = nt * 16 + lm;
            v8f acc = (v8f)0.0f;
            const int kkEnd = 4 * mt + 4;     // skip all-zero triangular K-tiles
            for (int kk = 0; kk < kkEnd; ++kk) {
                const int k0 = kk * 4 + kh;
                v2f a, bfr;
                a.x = (k0     <= m) ? 1.0f : 0.0f;
                a.y = (k0 + 1 <= m) ? 1.0f : 0.0f;
                bfr.x = lds[AOFF + (k0 + 1) * LDSTR + (n + 1)];
                bfr.y = lds[AOFF + (k0 + 2) * LDSTR + (n + 1)];
                acc = __builtin_amdgcn_wmma_f32_16x16x4_f32(false, a, false, bfr,
                                                            (short)0, acc, false, false);
            }
            const int mrow = mt * 16 + ((lane >> 4) << 3);
#pragma unroll
            for (int i = 0; i < 8; ++i)
                lds[BOFF + (mrow + i) * LDSTR + n] = acc[i];
        }
        __syncthreads();
        // GEMM-2: I = T * U  (col-cumsum over W). A = T from LDS (conflict-free col reads), B = U analytic.
        for (int t8 = 0; t8 < 8; ++t8) {
            const int mt = wv, nt = t8;       // work ~ nt -> balanced across waves
            const int m = mt * 16 + lm;
            const int n = nt * 16 + lm;
            v8f acc = (v8f)0.0f;
            const int kkEnd = 4 * nt + 4;
            for (int kk = 0; kk < kkEnd; ++kk) {
                const int k0 = kk * 4 + kh;
                v2f a, bfr;
                a.x = lds[BOFF + m * LDSTR + k0];
                a.y = lds[BOFF + m * LDSTR + k0 + 1];
                bfr.x = (k0     <= n) ? 1.0f : 0.0f;
                bfr.y = (k0 + 1 <= n) ? 1.0f : 0.0f;
                acc = __builtin_amdgcn_wmma_f32_16x16x4_f32(false, a, false, bfr,
                                                            (short)0, acc, false, false);
            }
            const int mrow = mt * 16 + ((lane >> 4) << 3);
#pragma unroll
            for (int i = 0; i < 8; ++i)
                lds[AOFF + (mrow + i + 1) * LDSTR + (n + 1)] = acc[i];
        }
    }
#else
    // Fallback: serial scans (no WMMA available)
    if (tid < Hd) {
        float s = 0.0f;
        for (int ww = 0; ww < Wd; ++ww) {
            s += lds[AOFF + (tid + 1) * LDSTR + (ww + 1)];
            lds[BOFF + tid * LDSTR + ww] = s;
        }
    }
    __syncthreads();
    if (tid < Wd) {
        float s = 0.0f;
        for (int hh = 0; hh < Hd; ++hh) {
            s += lds[BOFF + hh * LDSTR + tid];
            lds[AOFF + (hh + 1) * LDSTR + (tid + 1)] = s;
        }
    }
#endif
    __syncthreads();

    // ---- Phase 3: separable box interpolation; per f: E = col-interp(I), out = row-combine(E) ----
    const int cf0 = c * Fn;
    const int wq  = tid & 127;    // lane-major over w -> coalesced stores, conflict-free LDS
    const int seg = tid >> 7;
    for (int f = 0; f < Fn; ++f) {
        const float ym = y_min[cf0 + f] * 128.0f;
        const float yM = y_max[cf0 + f] * 128.0f;
        const float v0 = clip128((float)wq + ym);
        const int   j0 = imin((int)floorf(v0), 127);
        const float b0 = v0 - (float)j0;
        const float v1 = clip128((float)wq + yM + 1.0f);
        const int   j1 = imin((int)floorf(v1), 127);
        const float b1 = v1 - (float)j1;
        const int ib = seg ? 65 : 0;
        const int ie = seg ? 129 : 65;
        for (int i = ib; i < ie; ++i) {
            const int r = AOFF + i * LDSTR;
            const float e = (1.0f - b1) * lds[r + j1] + b1 * lds[r + j1 + 1]
                          - (1.0f - b0) * lds[r + j0] - b0 * lds[r + j0 + 1];
            lds[BOFF + i * LDSTR + wq] = e;
        }
        __syncthreads();

        const float xm = x_min[cf0 + f] * 128.0f;
        const float xM = x_max[cf0 + f] * 128.0f;
        float* __restrict__ op = out + ((size_t)bc * Fn + f) * (size_t)(Hd * Wd);
        for (int hq = seg * 64; hq < seg * 64 + 64; ++hq) {
            const float u0 = clip128((float)hq + xm);
            const int   i0 = imin((int)floorf(u0), 127);
            const float a0 = u0 - (float)i0;
            const float u1 = clip128((float)hq + xM + 1.0f);
            const int   i1 = imin((int)floorf(u1), 127);
            const float a1 = u1 - (float)i1;
            const float r = (1.0f - a1) * lds[BOFF + i1 * LDSTR + wq]
                          + a1 * lds[BOFF + (i1 + 1) * LDSTR + wq]
                          - (1.0f - a0) * lds[BOFF + i0 * LDSTR + wq]
                          - a0 * lds[BOFF + (i0 + 1) * LDSTR + wq];
            op[(size_t)hq * Wd + wq] = r;
        }
        __syncthreads();   // E region reused next f
    }
}

extern "C" void kernel_launch(void* const* d_in, const int* in_sizes, int n_in,
                              void* d_out, int out_size, void* d_ws, size_t ws_size,
                              hipStream_t stream) {
    (void)in_sizes; (void)n_in; (void)out_size; (void)d_ws; (void)ws_size;
    const float* x     = (const float*)d_in[0];
    const float* x_min = (const float*)d_in[1];
    const float* x_max = (const float*)d_in[2];
    const float* y_min = (const float*)d_in[3];
    const float* y_max = (const float*)d_in[4];
    float* out = (float*)d_out;

    const size_t shmem = (size_t)LDS_FLOATS * sizeof(float);   // 136,224 B (< 320 KB/WGP)
    (void)hipFuncSetAttribute((const void*)boxconv_fused,
                              hipFuncAttributeMaxDynamicSharedMemorySize, (int)shmem);
    boxconv_fused<<<dim3(Bn * Cn), dim3(256), shmem, stream>>>(x, x_min, x_max, y_min, y_max, out);
}